// OnlyLSTM_13142599926169
// MI455X (gfx1250) — compile-verified
//
#include <hip/hip_runtime.h>
#include <math.h>

// Problem constants (from reference)
#define BB 64      // batch
#define TT 512     // seq len
#define EE 256     // embed dim
#define HH 512     // hidden
#define NG 2048    // 4*H gate width
#define KK 768     // E + H fused K
#define NBLK 32    // persistent workgroups (one per 16-wide hidden block)
#define TPB 256    // 8 waves: 4 M-tiles x 2 K-halves

typedef float v2f __attribute__((ext_vector_type(2)));
typedef float v8f __attribute__((ext_vector_type(8)));

__device__ __forceinline__ float sigf(float x){ return 1.0f/(1.0f + __expf(-x)); }

// Pack [W_ih | W_hh] transposed + pair-interleaved:
//   Bw[(k>>1)*(2*NG) + n*2 + (k&1)] = W[n][k]
// so a lane's WMMA B fragment (k, k+1) at column n is one aligned float2 load.
__global__ void pack_weights(const float* __restrict__ Wih,
                             const float* __restrict__ Whh,
                             float* __restrict__ Bw){
  int i = blockIdx.x*blockDim.x + threadIdx.x;
  if (i >= KK*NG) return;
  int k = i / NG;
  int n = i - k*NG;
  float v = (k < EE) ? Wih[n*EE + k] : Whh[n*HH + (k - EE)];
  Bw[(size_t)(k >> 1)*(2*NG) + (size_t)n*2 + (k & 1)] = v;
}

__global__ void lstm_init(float* __restrict__ hbuf, unsigned* __restrict__ counter){
  unsigned i = blockIdx.x*blockDim.x + threadIdx.x;
  if (i == 0) *counter = 0u;
  for (unsigned j = i; j < BB*HH; j += gridDim.x*blockDim.x) hbuf[j] = 0.0f;
}

// Grid-wide phase barrier (single monotone counter; NBLK co-resident WGs).
__device__ __forceinline__ void gbar(unsigned* counter, unsigned phase){
  __syncthreads();
  if (threadIdx.x == 0){
    __hip_atomic_fetch_add(counter, 1u, __ATOMIC_ACQ_REL, __HIP_MEMORY_SCOPE_AGENT);
    while (__hip_atomic_load(counter, __ATOMIC_ACQUIRE, __HIP_MEMORY_SCOPE_AGENT)
           < phase*(unsigned)NBLK){
      __builtin_amdgcn_s_sleep(2);
    }
  }
  __syncthreads();
}

// One K=4 slab: A fragment (float2) vs 4 gate B fragments -> 4 WMMAs.
__device__ __forceinline__ void k4(v8f (&acc)[4], const float* aPtr,
                                   const float* bp, int n2){
  v2f af = *(const v2f*)aPtr;
  #pragma unroll
  for (int g = 0; g < 4; ++g){
    v2f bf = *(const v2f*)(bp + (size_t)g*(HH*2) + n2);
    acc[g] = __builtin_amdgcn_wmma_f32_16x16x4_f32(
        false, af, false, bf, (short)0, acc[g], false, false);
  }
}

__launch_bounds__(TPB, 1)
__global__ void lstm_persist(const int* __restrict__ x,
                             const int* __restrict__ length,
                             const float* __restrict__ emb,
                             const float* __restrict__ bih,
                             const float* __restrict__ bhh,
                             const float* __restrict__ Wcls,
                             const float* __restrict__ bcls,
                             const float* __restrict__ Bw,
                             float* __restrict__ hbuf,     // 2 * [BB][HH] double buffer
                             float* __restrict__ feat,     // [BB][HH]
                             unsigned* __restrict__ counter,
                             float* __restrict__ out)      // [BB][2]
{
  __shared__ float red[4][4][8][32];   // kh=1 partial C: [mt][gate][e][lane], 16KB

  const int tid  = threadIdx.x;
  const int wave = tid >> 5;
  const int lane = tid & 31;
  const int l15  = lane & 15;
  const int lh   = lane >> 4;          // half-wave: A/B K offset +2
  const int mt   = wave & 3;           // M-tile (16 batch rows)
  const int kh   = wave >> 2;          // K-half: 0 -> emb + h[0,128); 1 -> h[128,512)
  const int hb   = blockIdx.x;         // hidden block 0..31 (16 cols)
  const int ncol = hb*16 + l15;        // this lane's hidden column in C/D
  const int rowA = mt*16 + l15;        // this lane's A-fragment row (batch index)
  const int n2   = ncol*2;             // Bw inner offset for gate 0

  // Per-gate bias (same for all rows of a column); folded into kh=0 C init.
  float bias[4];
  #pragma unroll
  for (int g = 0; g < 4; ++g) bias[g] = bih[g*HH + ncol] + bhh[g*HH + ncol];

  // Lengths for the 8 C rows this lane owns (C layout: M = vgpr + 8*(lane>=16)).
  int lenv[8];
  #pragma unroll
  for (int e = 0; e < 8; ++e) lenv[e] = length[mt*16 + e + 8*lh];

  v8f cc, mf;   // cell state + running max, VGPR-resident all 512 steps
  #pragma unroll
  for (int e = 0; e < 8; ++e){ cc[e] = 0.0f; mf[e] = -3.402823466e38f; }

  for (int t = 0; t < TT; ++t){
    gbar(counter, (unsigned)(t+1));    // h_t fully published by all WGs

    const float* hprev = hbuf + (size_t)(t & 1)*BB*HH;
    const float* aH    = hprev + (size_t)rowA*HH;

    v8f acc[4];
    #pragma unroll
    for (int g = 0; g < 4; ++g)
      #pragma unroll
      for (int e = 0; e < 8; ++e) acc[g][e] = (kh == 0) ? bias[g] : 0.0f;

    if (kh == 0){
      // Embedding part of K (gathered rows; PAD row of table is zero).
      const int   tok  = x[rowA*TT + t];
      const float* aE  = emb + (size_t)tok*EE;
      #pragma unroll 2
      for (int k0 = 0; k0 < EE; k0 += 4){
        const int kk = k0 + 2*lh;
        k4(acc, aE + kk, Bw + (size_t)(kk >> 1)*(2*NG), n2);
      }
      // Recurrent part, k in [0,128)
      #pragma unroll 2
      for (int k0 = 0; k0 < 128; k0 += 4){
        const int kk = k0 + 2*lh;
        k4(acc, aH + kk, Bw + (size_t)((kk + EE) >> 1)*(2*NG), n2);
      }
    } else {
      // Recurrent part, k in [128,512)
      #pragma unroll 2
      for (int k0 = 128; k0 < HH; k0 += 4){
        const int kk = k0 + 2*lh;
        k4(acc, aH + kk, Bw + (size_t)((kk + EE) >> 1)*(2*NG), n2);
      }
    }

    if (kh == 1){
      #pragma unroll
      for (int g = 0; g < 4; ++g)
        #pragma unroll
        for (int e = 0; e < 8; ++e) red[mt][g][e][lane] = acc[g][e];
    }
    __syncthreads();

    if (kh == 0){
      #pragma unroll
      for (int g = 0; g < 4; ++g)
        #pragma unroll
        for (int e = 0; e < 8; ++e) acc[g][e] += red[mt][g][e][lane];

      // Gate nonlinearities + state update + ragged max-pool (order: i,f,g,o).
      float* hnext = hbuf + (size_t)((t+1) & 1)*BB*HH;
      #pragma unroll
      for (int e = 0; e < 8; ++e){
        int row = mt*16 + e + 8*lh;
        float ig = sigf(acc[0][e]);
        float fg = sigf(acc[1][e]);
        float gg = tanhf(acc[2][e]);
        float og = sigf(acc[3][e]);
        float c  = fg*cc[e] + ig*gg;
        cc[e] = c;
        float h  = og*tanhf(c);
        hnext[(size_t)row*HH + ncol] = h;
        float cand = (t < lenv[e]) ? h : -3.402823466e38f;
        mf[e] = fmaxf(mf[e], cand);
      }
    }
  }

  // Publish pooled features (kh=0 waves own the final state).
  if (kh == 0){
    #pragma unroll
    for (int e = 0; e < 8; ++e){
      int row = mt*16 + e + 8*lh;
      feat[(size_t)row*HH + ncol] = mf[e];
    }
  }

  gbar(counter, (unsigned)(TT + 1));

  // Tiny classifier: 64x2, done by block 0.
  if (blockIdx.x == 0){
    for (int i = tid; i < BB*2; i += TPB){
      int b = i >> 1, c = i & 1;
      float s = bcls[c];
      for (int j = 0; j < HH; ++j) s += feat[(size_t)b*HH + j]*Wcls[c*HH + j];
      out[b*2 + c] = s;
    }
  }
}

extern "C" void kernel_launch(void* const* d_in, const int* in_sizes, int n_in,
                              void* d_out, int out_size, void* d_ws, size_t ws_size,
                              hipStream_t stream) {
  (void)in_sizes; (void)n_in; (void)out_size; (void)ws_size;
  const int*   xids  = (const int*)  d_in[0];   // [B,T] int32
  const int*   len   = (const int*)  d_in[1];   // [B,1] int32
  const float* emb   = (const float*)d_in[2];   // [32000,256]
  const float* Wih   = (const float*)d_in[3];   // [2048,256]
  const float* Whh   = (const float*)d_in[4];   // [2048,512]
  const float* bih   = (const float*)d_in[5];   // [2048]
  const float* bhh   = (const float*)d_in[6];   // [2048]
  const float* Wcls  = (const float*)d_in[7];   // [2,512]
  const float* bcls  = (const float*)d_in[8];   // [2]
  float* out = (float*)d_out;                   // [64,2]

  // Workspace layout (~6.7 MB total)
  char* ws = (char*)d_ws;
  float* Bw   = (float*)ws;                               // KK*NG floats (6 MB)
  float* hbuf = (float*)(ws + (size_t)KK*NG*4);           // 2*BB*HH floats
  float* feat = hbuf + 2*BB*HH;                           // BB*HH floats
  unsigned* counter = (unsigned*)(feat + BB*HH);

  pack_weights<<<(KK*NG + 255)/256, 256, 0, stream>>>(Wih, Whh, Bw);
  lstm_init<<<32, 256, 0, stream>>>(hbuf, counter);
  lstm_persist<<<NBLK, TPB, 0, stream>>>(xids, len, emb, bih, bhh, Wcls, bcls,
                                         Bw, hbuf, feat, counter, out);
}